// Quantizer_85435489452138
// MI455X (gfx1250) — compile-verified
//
#include <hip/hip_runtime.h>
#include <hip/hip_bf16.h>

// Soft-quantizer forward on gfx1250 (MI455X).
//
// Key algebraic fact: the straight-through estimator's forward value is
//   z_bar = z_soft + (z_hard - z_soft) == z_hard = centers[argmin_q (z-c_q)^2]
// (exact in reals; <= ~2e-7 abs difference in fp32), so the softmax is dead
// code for the validated forward outputs. We compute only the argmin and the
// hard codebook lookup, which takes the kernel from VALU-bound to ~balanced
// against the 23.3 TB/s HBM floor (~100 MB of traffic => ~4.3 us).
//
// argmin_q (z-c_q)^2 == argmax_q (2*c_q*z - c_q^2): 1 FMA + cmp + 2 selects
// per center, lane-local (no cross-lane traffic, so WMMA would only add
// shuffle overhead). Streaming data uses non-temporal hints so 100 MB of
// single-use data does not churn the 192 MB L2; the 64 B codebook is wave-
// uniform and lands in SGPRs via scalar loads.

typedef float v4f __attribute__((ext_vector_type(4)));
typedef int   v4i __attribute__((ext_vector_type(4)));

#define QC  16   // codebook size
#define EPT 16   // elements per thread = 4 x b128
#define BLK 256  // 8 waves per workgroup on wave32

__global__ __launch_bounds__(BLK) void quantizer_fwd_kernel(
    const float* __restrict__ z,
    const float* __restrict__ centers,
    float*       __restrict__ zbar,
    int*         __restrict__ indices,
    int n)
{
    // Wave-uniform codebook -> scalar loads; coefficients for the monotone
    // score l_q = 2*c_q*z - c_q^2 (same argmax as argmin of (z-c_q)^2).
    float a[QC], b[QC];
#pragma unroll
    for (int q = 0; q < QC; ++q) {
        float c = centers[q];
        a[q] = 2.0f * c;
        b[q] = -c * c;
    }

    const long long base = (long long)(blockIdx.x * BLK + threadIdx.x) * EPT;
    if (base >= n) return;

    if (base + EPT <= n) {
        const v4f* zp = (const v4f*)(z + base);
        // Pull the stream ahead into cache (gfx1250 global_prefetch_b8).
        __builtin_prefetch((const char*)zp + 16384, 0, 0);

        v4f zv[EPT / 4];
#pragma unroll
        for (int v = 0; v < EPT / 4; ++v)
            zv[v] = __builtin_nontemporal_load(zp + v);

        v4f hard[EPT / 4];
        v4i idx[EPT / 4];
#pragma unroll
        for (int v = 0; v < EPT / 4; ++v) {
#pragma unroll
            for (int e = 0; e < 4; ++e) {
                const float zz = zv[v][e];
                float best = __builtin_fmaf(a[0], zz, b[0]);
                int   bi   = 0;
#pragma unroll
                for (int q = 1; q < QC; ++q) {
                    const float l  = __builtin_fmaf(a[q], zz, b[q]);
                    const bool  gt = l > best;          // strict: keep FIRST max
                    best = gt ? l : best;               // == first-min of dist
                    bi   = gt ? q : bi;
                }
                idx[v][e]  = bi;
                hard[v][e] = centers[bi];               // 64 B table, cache-hot gather
            }
        }

#pragma unroll
        for (int v = 0; v < EPT / 4; ++v) {
            __builtin_nontemporal_store(hard[v], (v4f*)(zbar + base) + v);
            __builtin_nontemporal_store(idx[v],  (v4i*)(indices + base) + v);
        }
    } else {
        // Scalar tail (not hit for 8,388,608 elements, kept for generality).
        for (long long i = base; i < n; ++i) {
            const float zz = z[i];
            float best = __builtin_fmaf(a[0], zz, b[0]);
            int   bi   = 0;
#pragma unroll
            for (int q = 1; q < QC; ++q) {
                const float l  = __builtin_fmaf(a[q], zz, b[q]);
                const bool  gt = l > best;
                best = gt ? l : best;
                bi   = gt ? q : bi;
            }
            zbar[i]    = centers[bi];
            indices[i] = bi;
        }
    }
}

extern "C" void kernel_launch(void* const* d_in, const int* in_sizes, int n_in,
                              void* d_out, int out_size, void* d_ws, size_t ws_size,
                              hipStream_t stream) {
    const float* z       = (const float*)d_in[0];
    const float* centers = (const float*)d_in[1];
    const int    n       = in_sizes[0];      // 16*32*16384 = 8,388,608

    float* zbar    = (float*)d_out;          // first n floats: z_bar (== z_hard fwd)
    int*   indices = (int*)d_out + n;        // next n int32:   argmin indices

    const int threads = (n + EPT - 1) / EPT;
    const int blocks  = (threads + BLK - 1) / BLK;
    quantizer_fwd_kernel<<<blocks, BLK, 0, stream>>>(z, centers, zbar, indices, n);
}